// Model_4312147165864
// MI455X (gfx1250) — compile-verified
//
#include <hip/hip_runtime.h>
#include <math.h>

#define N_NODES 200000
#define N_EDGES 3200000
#define N_FEAT  128
#define N_GRAPHS 1000
#define KTOP    30
#define C_CAT   97

typedef __attribute__((ext_vector_type(2))) float v2f;
typedef __attribute__((ext_vector_type(8))) float v8f;

// ---------------------------------------------------------------- utilities
__global__ void fill_f32(float* p, float v, long n) {
    long i = (long)blockIdx.x * blockDim.x + threadIdx.x;
    if (i < n) p[i] = v;
}

// ---------------------------------------------------------------- degrees
__global__ void deg_accum(const int* __restrict__ src, const int* __restrict__ dst,
                          float* deg, int E) {
    int e = blockIdx.x * blockDim.x + threadIdx.x;
    if (e >= E) return;
    int s = src[e], d = dst[e];
    if (s != d) unsafeAtomicAdd(&deg[d], 1.0f);
}

__global__ void deg_finalize(float* inv_deg /*in: deg*/, float* inv_sqrt, int N) {
    int n = blockIdx.x * blockDim.x + threadIdx.x;
    if (n >= N) return;
    float d = inv_deg[n] + 1.0f;          // + self loop
    inv_sqrt[n] = rsqrtf(d);
    inv_deg[n]  = 1.0f / d;
}

// ---------------------------------------------------------------- WMMA GEMM
// out[rows x 32] = A[rows x K] @ W[K x 32]; rows % 16 == 0.
// One wave per 16-row tile, two 16x16 n-tiles held in two f32 accumulators.
// V_WMMA_F32_16X16X4_F32: A 16x4 (lane&15=M, lane>>4 picks K pair),
// B 4x16 (lane&15=N), C/D 8 VGPRs (lanes16-31 -> M+8). Full fp32 precision.
__global__ void gemm_wmma_n32(const float* __restrict__ A, int lda,
                              const float* __restrict__ W, int K,
                              float* __restrict__ out, int rows) {
    int gt   = blockIdx.x * blockDim.x + threadIdx.x;
    int wave = gt >> 5;                    // uniform within wave
    int lane = threadIdx.x & 31;
    int row_base = wave * 16;
    if (row_base >= rows) return;          // wave-uniform: EXEC all-1 past here
    int l15  = lane & 15;
    int half = lane >> 4;

    v8f acc0 = {};
    v8f acc1 = {};
    const float* arow = A + (long)(row_base + l15) * lda;
    for (int k0 = 0; k0 < K; k0 += 4) {
        int ka = k0 + 2 * half;
        v2f a;  a[0] = arow[ka];                a[1] = arow[ka + 1];
        v2f b0; b0[0] = W[ka * 32 + l15];       b0[1] = W[(ka + 1) * 32 + l15];
        v2f b1; b1[0] = W[ka * 32 + 16 + l15];  b1[1] = W[(ka + 1) * 32 + 16 + l15];
        acc0 = __builtin_amdgcn_wmma_f32_16x16x4_f32(false, a, false, b0, (short)0, acc0, false, false);
        acc1 = __builtin_amdgcn_wmma_f32_16x16x4_f32(false, a, false, b1, (short)0, acc1, false, false);
    }
    for (int v = 0; v < 8; ++v) {
        int row = row_base + v + 8 * half;
        out[(long)row * 32 + l15]      = acc0[v];
        out[(long)row * 32 + 16 + l15] = acc1[v];
    }
}

// ---------------------------------------------------------------- GCN pieces
__global__ void init_agg32(const float* __restrict__ hw, const float* __restrict__ inv_deg,
                           const float* __restrict__ b, float* agg, int N) {
    int t = blockIdx.x * blockDim.x + threadIdx.x;
    if (t >= N * 32) return;
    int n = t >> 5, c = t & 31;
    agg[t] = b[c] + hw[t] * inv_deg[n];    // self-loop + bias
}

// one wave per edge, one lane per channel (32 channels == wave32)
__global__ void scatter32(const int* __restrict__ src, const int* __restrict__ dst,
                          const float* __restrict__ inv_sqrt,
                          const float* __restrict__ hw, float* agg, int E) {
    int gt = blockIdx.x * blockDim.x + threadIdx.x;
    int e = gt >> 5;
    int lane = threadIdx.x & 31;
    if (e >= E) return;
    int s = src[e], d = dst[e];
    if (s == d) return;                    // removed self loops
    float coef = inv_sqrt[s] * inv_sqrt[d];
    float v = hw[(long)s * 32 + lane] * coef;
    unsafeAtomicAdd(&agg[(long)d * 32 + lane], v);
}

__global__ void tanh_to_cat(const float* __restrict__ agg, float* cat, int coff, int N) {
    int t = blockIdx.x * blockDim.x + threadIdx.x;
    if (t >= N * 32) return;
    int n = t >> 5, c = t & 31;
    cat[(long)n * C_CAT + coff + c] = tanhf(agg[t]);
}

// layer 4: 32 -> 1 channel
__global__ void gcn4_hw(const float* __restrict__ cat, const float* __restrict__ W4,
                        float* hw1, int N) {
    int n = blockIdx.x * blockDim.x + threadIdx.x;
    if (n >= N) return;
    const float* row = cat + (long)n * C_CAT + 64;
    float s = 0.f;
    for (int c = 0; c < 32; ++c) s += row[c] * W4[c];
    hw1[n] = s;
}
__global__ void init_agg1(const float* __restrict__ hw1, const float* __restrict__ inv_deg,
                          const float* __restrict__ b4, float* agg1, int N) {
    int n = blockIdx.x * blockDim.x + threadIdx.x;
    if (n >= N) return;
    agg1[n] = b4[0] + hw1[n] * inv_deg[n];
}
__global__ void scatter1(const int* __restrict__ src, const int* __restrict__ dst,
                         const float* __restrict__ inv_sqrt,
                         const float* __restrict__ hw1, float* agg1, int E) {
    int e = blockIdx.x * blockDim.x + threadIdx.x;
    if (e >= E) return;
    int s = src[e], d = dst[e];
    if (s == d) return;
    unsafeAtomicAdd(&agg1[d], hw1[s] * inv_sqrt[s] * inv_sqrt[d]);
}
__global__ void tanh4_to_cat(const float* __restrict__ agg1, float* cat, int N) {
    int n = blockIdx.x * blockDim.x + threadIdx.x;
    if (n >= N) return;
    cat[(long)n * C_CAT + 96] = tanhf(agg1[n]);
}

// ---------------------------------------------------------------- sort pool
__global__ void count_nodes(const int* __restrict__ batch, int* counts, int N) {
    int n = blockIdx.x * blockDim.x + threadIdx.x;
    if (n >= N) return;
    atomicAdd(&counts[batch[n]], 1);
}
__global__ void scan_starts(const int* __restrict__ counts, int* starts, int G) {
    if (blockIdx.x == 0 && threadIdx.x == 0) {
        int acc = 0;
        for (int g = 0; g < G; ++g) { starts[g] = acc; acc += counts[g]; }
    }
}

#define MAXSH 2048
__global__ void sortpool(const float* __restrict__ cat, const int* __restrict__ counts,
                         const int* __restrict__ starts, float* pooled) {
    __shared__ float vals[MAXSH];
    int g = blockIdx.x;
    int cnt = counts[g], st = starts[g];
    int nsh = cnt < MAXSH ? cnt : MAXSH;
    for (int i = threadIdx.x; i < nsh; i += blockDim.x)
        vals[i] = cat[(long)(st + i) * C_CAT + 96];
    __syncthreads();
    for (int i = threadIdx.x; i < cnt; i += blockDim.x) {
        float v = (i < nsh) ? vals[i] : cat[(long)(st + i) * C_CAT + 96];
        int rank = 0;
        for (int j = 0; j < cnt; ++j) {
            float vj = (j < nsh) ? vals[j] : cat[(long)(st + j) * C_CAT + 96];
            rank += (vj > v) || (vj == v && j < i);   // stable desc (lexsort)
        }
        if (rank < KTOP) {
            const float* sr = cat + (long)(st + i) * C_CAT;
            float* dr = pooled + ((long)g * KTOP + rank) * C_CAT;
            for (int c = 0; c < C_CAT; ++c) dr[c] = sr[c];
        }
    }
}

// ------------------------------------------- conv5 (k=97,s=97) + relu + maxpool2
__global__ void conv5_pool(const float* __restrict__ pooled, const float* __restrict__ w5,
                           const float* __restrict__ b5, float* z5) {
    __shared__ float sp[KTOP * C_CAT];
    __shared__ float sw[16 * C_CAT];
    int g = blockIdx.x;
    for (int i = threadIdx.x; i < KTOP * C_CAT; i += blockDim.x)
        sp[i] = pooled[(long)g * KTOP * C_CAT + i];
    for (int i = threadIdx.x; i < 16 * C_CAT; i += blockDim.x) sw[i] = w5[i];
    __syncthreads();
    for (int t = threadIdx.x; t < 16 * 15; t += blockDim.x) {
        int o = t / 15, p2 = t % 15;
        float m = -INFINITY;
        for (int h = 0; h < 2; ++h) {
            int p = 2 * p2 + h;
            float s = b5[o];
            for (int c = 0; c < C_CAT; ++c) s += sp[p * C_CAT + c] * sw[o * C_CAT + c];
            m = fmaxf(m, fmaxf(s, 0.f));
        }
        z5[((long)g * 16 + o) * 15 + p2] = m;
    }
}

// --------------------------------- conv6 + relu + fc1 + relu + fc2 + log_softmax
__global__ void head(const float* __restrict__ z5, const float* __restrict__ w6,
                     const float* __restrict__ b6, const float* __restrict__ fc1w,
                     const float* __restrict__ fc1b, const float* __restrict__ fc2w,
                     const float* __restrict__ fc2b, float* __restrict__ out) {
    __shared__ float s5[16 * 15];
    __shared__ float s6[352];
    __shared__ float sh[128];
    __shared__ float slog[10];
    __shared__ float lse;
    int g = blockIdx.x;
    for (int i = threadIdx.x; i < 240; i += blockDim.x) s5[i] = z5[(long)g * 240 + i];
    __syncthreads();
    for (int t = threadIdx.x; t < 352; t += blockDim.x) {       // conv6: [32 x 11]
        int o = t / 11, p = t % 11;
        float s = b6[o];
        for (int i = 0; i < 16; ++i)
            for (int k = 0; k < 5; ++k)
                s += s5[i * 15 + p + k] * w6[o * 80 + i * 5 + k];
        s6[t] = fmaxf(s, 0.f);
    }
    __syncthreads();
    int j = threadIdx.x;
    if (j < 128) {                                              // fc1
        float s = fc1b[j];
        for (int v = 0; v < 352; ++v) s += s6[v] * fc1w[v * 128 + j];
        sh[j] = fmaxf(s, 0.f);
    }
    __syncthreads();
    if (j < 10) {                                               // fc2
        float s = fc2b[j];
        for (int v = 0; v < 128; ++v) s += sh[v] * fc2w[v * 10 + j];
        slog[j] = s;
    }
    __syncthreads();
    if (j == 0) {                                               // log-softmax
        float mx = slog[0];
        for (int k = 1; k < 10; ++k) mx = fmaxf(mx, slog[k]);
        float se = 0.f;
        for (int k = 0; k < 10; ++k) se += expf(slog[k] - mx);
        lse = mx + logf(se);
    }
    __syncthreads();
    if (j < 10) out[(long)g * 10 + j] = slog[j] - lse;
}

// ---------------------------------------------------------------- launcher
extern "C" void kernel_launch(void* const* d_in, const int* in_sizes, int n_in,
                              void* d_out, int out_size, void* d_ws, size_t ws_size,
                              hipStream_t stream) {
    const float* x     = (const float*)d_in[0];
    const int*   ei    = (const int*)d_in[1];
    const int*   srcE  = ei;
    const int*   dstE  = ei + N_EDGES;
    const int*   batch = (const int*)d_in[2];
    const float* W1 = (const float*)d_in[3],  *b1 = (const float*)d_in[4];
    const float* W2 = (const float*)d_in[5],  *b2 = (const float*)d_in[6];
    const float* W3 = (const float*)d_in[7],  *b3 = (const float*)d_in[8];
    const float* W4 = (const float*)d_in[9],  *b4 = (const float*)d_in[10];
    const float* c5w = (const float*)d_in[11], *c5b = (const float*)d_in[12];
    const float* c6w = (const float*)d_in[13], *c6b = (const float*)d_in[14];
    const float* f1w = (const float*)d_in[15], *f1b = (const float*)d_in[16];
    const float* f2w = (const float*)d_in[17], *f2b = (const float*)d_in[18];
    float* out = (float*)d_out;

    const int N = N_NODES, E = N_EDGES, G = N_GRAPHS;

    // workspace layout (floats)
    float* ws       = (float*)d_ws;
    float* inv_deg  = ws;                         // N (also deg accumulator)
    float* inv_sqrt = inv_deg + N;                // N
    float* hw       = inv_sqrt + N;               // N*32
    float* agg      = hw + (long)N * 32;          // N*32
    float* cat      = agg + (long)N * 32;         // N*97
    float* pooled   = cat + (long)N * C_CAT;      // G*30*97
    float* z5       = pooled + (long)G * KTOP * C_CAT; // G*240
    int*   counts   = (int*)(z5 + (long)G * 240); // G
    int*   starts   = counts + G;                 // G

    const int T = 256;
    auto blk = [](long n, int t) { return (int)((n + t - 1) / t); };

    // --- degrees
    fill_f32<<<blk(N, T), T, 0, stream>>>(inv_deg, 0.f, N);
    deg_accum<<<blk(E, T), T, 0, stream>>>(srcE, dstE, inv_deg, E);
    deg_finalize<<<blk(N, T), T, 0, stream>>>(inv_deg, inv_sqrt, N);

    // --- GCN layers (GEMM waves: 16 rows each; 320 thr = 10 waves/block)
    int gemm_blocks = blk((long)(N / 16) * 32, 320);
    // layer 1: x[ N x 128 ] @ W1
    gemm_wmma_n32<<<gemm_blocks, 320, 0, stream>>>(x, N_FEAT, W1, N_FEAT, hw, N);
    init_agg32<<<blk((long)N * 32, T), T, 0, stream>>>(hw, inv_deg, b1, agg, N);
    scatter32<<<blk((long)E * 32, T), T, 0, stream>>>(srcE, dstE, inv_sqrt, hw, agg, E);
    tanh_to_cat<<<blk((long)N * 32, T), T, 0, stream>>>(agg, cat, 0, N);
    // layer 2: cat[:,0:32] @ W2
    gemm_wmma_n32<<<gemm_blocks, 320, 0, stream>>>(cat + 0, C_CAT, W2, 32, hw, N);
    init_agg32<<<blk((long)N * 32, T), T, 0, stream>>>(hw, inv_deg, b2, agg, N);
    scatter32<<<blk((long)E * 32, T), T, 0, stream>>>(srcE, dstE, inv_sqrt, hw, agg, E);
    tanh_to_cat<<<blk((long)N * 32, T), T, 0, stream>>>(agg, cat, 32, N);
    // layer 3: cat[:,32:64] @ W3
    gemm_wmma_n32<<<gemm_blocks, 320, 0, stream>>>(cat + 32, C_CAT, W3, 32, hw, N);
    init_agg32<<<blk((long)N * 32, T), T, 0, stream>>>(hw, inv_deg, b3, agg, N);
    scatter32<<<blk((long)E * 32, T), T, 0, stream>>>(srcE, dstE, inv_sqrt, hw, agg, E);
    tanh_to_cat<<<blk((long)N * 32, T), T, 0, stream>>>(agg, cat, 64, N);
    // layer 4: cat[:,64:96] @ W4 (32 -> 1)
    gcn4_hw<<<blk(N, T), T, 0, stream>>>(cat, W4, hw, N);
    init_agg1<<<blk(N, T), T, 0, stream>>>(hw, inv_deg, b4, agg, N);
    scatter1<<<blk(E, T), T, 0, stream>>>(srcE, dstE, inv_sqrt, hw, agg, E);
    tanh4_to_cat<<<blk(N, T), T, 0, stream>>>(agg, cat, N);

    // --- sort pooling
    fill_f32<<<blk((long)G * KTOP * C_CAT, T), T, 0, stream>>>(pooled, 0.f, (long)G * KTOP * C_CAT);
    fill_f32<<<blk(G, T), T, 0, stream>>>((float*)counts, 0.f, G); // zero bits
    count_nodes<<<blk(N, T), T, 0, stream>>>(batch, counts, N);
    scan_starts<<<1, 32, 0, stream>>>(counts, starts, G);
    sortpool<<<G, 256, 0, stream>>>(cat, counts, starts, pooled);

    // --- conv head
    conv5_pool<<<G, 256, 0, stream>>>(pooled, c5w, c5b, z5);
    head<<<G, 128, 0, stream>>>(z5, c6w, c6b, f1w, f1b, f2w, f2b, out);
}